// PagedAttention_31104153157682
// MI455X (gfx1250) — compile-verified
//
#include <hip/hip_runtime.h>

// Paged attention decode (GQA), fp32, MI455X / gfx1250 (wave32, WMMA).
// B=64, H=32, KV=8, D=128, BS=16, MB=128, S=2048, G=H/KV=4.
// Memory-bound: ~0.8 GB of K/V traffic -> ~35us floor at 23.3 TB/s.

#define BQ   64
#define HQ   32
#define KVH  8
#define DH   128
#define BSZ  16
#define MBL  128
#define GQ   4
#define NWAVE 8
#define NEGINF (-1e30f)

typedef float v2f __attribute__((ext_vector_type(2)));
typedef float v8f __attribute__((ext_vector_type(8)));

__device__ __forceinline__ float xor16(float x) {
    return __shfl_xor(x, 16, 32);
}

__global__ __launch_bounds__(256)
void paged_attn_decode_kernel(const float* __restrict__ q,
                              const float* __restrict__ kf,
                              const float* __restrict__ vf,
                              const float* __restrict__ kc,
                              const float* __restrict__ vc,
                              const int*   __restrict__ ctx_lens,
                              const int*   __restrict__ btab,
                              float*       __restrict__ out)
{
    const int kv   = blockIdx.x;          // kv head
    const int b    = blockIdx.y;          // sequence
    const int tid  = threadIdx.x;
    const int lane = tid & 31;
    const int wave = tid >> 5;
    const int lh   = lane & 15;           // head column (N) / row-in-half (M)
    const bool upper = lane >= 16;
    const int koff = upper ? 2 : 0;       // K-column offset inside 4-chunk

    __shared__ float red_m[NWAVE * 16];
    __shared__ float red_l[NWAVE * 16];
    __shared__ float outbuf[DH * 16];     // O^T [d][head], fp32 accumulate

    for (int i = tid; i < DH * 16; i += 256) outbuf[i] = 0.0f;
    __syncthreads();

    const int   ctx   = ctx_lens[b];
    const float scale = 0.08838834764831845f;   // 1/sqrt(128)

    // ---- Q^T fragments (B operand of QK wmma), held in registers for whole loop.
    // B[k][n] = Q[head n][4c + k];  per lane: .x = Q[lh][4c+koff], .y = +1
    v2f qfrag[32];
    {
        const float* qrow = q + ((size_t)b * HQ + (size_t)kv * GQ + lh) * DH;
        #pragma unroll
        for (int c = 0; c < 32; ++c) {
            if (lh < GQ) {
                qfrag[c].x = qrow[4 * c + koff];
                qfrag[c].y = qrow[4 * c + koff + 1];
            } else {
                qfrag[c].x = 0.0f;
                qfrag[c].y = 0.0f;
            }
        }
    }

    v8f acc[8];                            // O^T accumulators: acc[dt][r] = O^T[dt*16+r(+8)][lh]
    #pragma unroll
    for (int i = 0; i < 8; ++i) acc[i] = (v8f){0.f,0.f,0.f,0.f,0.f,0.f,0.f,0.f};

    float m = NEGINF, l = 0.0f;

    const float* kfresh = kf + ((size_t)b * KVH + kv) * DH;
    const float* vfresh = vf + ((size_t)b * KVH + kv) * DH;
    const size_t rowstride = (size_t)KVH * DH;   // slot stride inside a cache block

    const int ntiles = (ctx + BSZ - 1) / BSZ;    // one tile == one cache block (BS==16)

    for (int t = wave; t < ntiles; t += NWAVE) {
        const int    blk      = btab[b * MBL + t];
        const size_t tilebase = (((size_t)blk * BSZ) * KVH + kv) * DH;
        const int    freshRow = ctx - 1 - t * BSZ;   // in [0,15] only for the decode tile

        // Prefetch next tile's K block (CDNA5 global_prefetch path).
        if (t + NWAVE < ntiles) {
            const int nblk = btab[b * MBL + t + NWAVE];
            const size_t nbase = (((size_t)nblk * BSZ) * KVH + kv) * DH;
            __builtin_prefetch(kc + nbase + (size_t)lh * rowstride, 0, 1);
        }

        // ---- scores^T = K_tile x Q^T via V_WMMA_F32_16X16X4_F32 (32 chunks over D=128)
        // A[m][k] = K[key m][4c+k]; lane handles key row lh, columns koff/koff+1.
        // Two independent accumulation chains (even/odd chunks) for 2x WMMA ILP:
        // a serial 32-deep C-accumulation chain throttles a single wave, and with
        // only 2 waves/SIMD there is not enough cross-wave work to hide it.
        const float* kptr = (lh == freshRow)
                          ? kfresh
                          : (kc + tilebase + (size_t)lh * rowstride);
        v8f s0 = (v8f){0.f,0.f,0.f,0.f,0.f,0.f,0.f,0.f};
        v8f s1 = (v8f){0.f,0.f,0.f,0.f,0.f,0.f,0.f,0.f};
        #pragma unroll
        for (int c = 0; c < 16; ++c) {
            v2f a0, a1;
            a0.x = kptr[8 * c + koff];
            a0.y = kptr[8 * c + koff + 1];
            a1.x = kptr[8 * c + 4 + koff];
            a1.y = kptr[8 * c + 4 + koff + 1];
            s0 = __builtin_amdgcn_wmma_f32_16x16x4_f32(
                     false, a0, false, qfrag[2 * c],     (short)0, s0, false, false);
            s1 = __builtin_amdgcn_wmma_f32_16x16x4_f32(
                     false, a1, false, qfrag[2 * c + 1], (short)0, s1, false, false);
        }
        const v8f s = s0 + s1;

        // ---- scale + mask + online softmax (per head column = per lane)
        float p[8];
        float tmax = NEGINF;
        #pragma unroll
        for (int r = 0; r < 8; ++r) {
            const int key = t * BSZ + r + (upper ? 8 : 0);
            const float sv = (key < ctx) ? s[r] * scale : NEGINF;
            p[r] = sv;
            tmax = fmaxf(tmax, sv);
        }
        tmax = fmaxf(tmax, xor16(tmax));          // combine the two key-halves
        const float mnew  = fmaxf(m, tmax);
        const float alpha = __expf(m - mnew);
        float rsum = 0.0f;
        #pragma unroll
        for (int r = 0; r < 8; ++r) {
            p[r] = __expf(p[r] - mnew);           // masked -> exp(NEG)=0
            rsum += p[r];
        }
        rsum += xor16(rsum);
        l = l * alpha + rsum;
        m = mnew;

        #pragma unroll
        for (int i = 0; i < 8; ++i) acc[i] = acc[i] * alpha;

        // ---- O^T += V^T x P^T  (4 key-chunks x 8 d-tiles of WMMA)
        // dt-inner order: consecutive WMMAs target distinct acc registers (8-way ILP).
        #pragma unroll
        for (int sc = 0; sc < 4; ++sc) {
            const int  pb = (sc & 1) ? 4 : 0;     // keys 4sc..4sc+3 live in p[pb..pb+3]
            const bool hi = (sc >= 2);            // upper-half keys (8..15)

            // B2 fragment (P^T chunk): half-wave swap via xor-16 shuffle + select.
            const float t0 = xor16(p[pb + (hi ? 0 : 2)]);
            const float t1 = xor16(p[pb + (hi ? 1 : 3)]);
            v2f b2;
            if (hi) {
                b2.x = (!upper) ? t0 : p[pb + 2];
                b2.y = (!upper) ? t1 : p[pb + 3];
            } else {
                b2.x = (!upper) ? p[pb + 0] : t0;
                b2.y = (!upper) ? p[pb + 1] : t1;
            }

            // A2 rows: V rows 4sc+koff, 4sc+koff+1 of this tile (fresh substitute).
            const int r0 = 4 * sc + koff;
            const int r1 = r0 + 1;
            const float* vr0 = (r0 == freshRow) ? vfresh
                               : (vc + tilebase + (size_t)r0 * rowstride);
            const float* vr1 = (r1 == freshRow) ? vfresh
                               : (vc + tilebase + (size_t)r1 * rowstride);

            #pragma unroll
            for (int dt = 0; dt < 8; ++dt) {
                v2f a2;
                a2.x = vr0[dt * 16 + lh];
                a2.y = vr1[dt * 16 + lh];
                acc[dt] = __builtin_amdgcn_wmma_f32_16x16x4_f32(
                              false, a2, false, b2, (short)0, acc[dt], false, false);
            }
        }
    }

    // ---- cross-wave flash-decode merge through LDS
    if (lane < 16) {
        red_m[wave * 16 + lane] = m;
        red_l[wave * 16 + lane] = l;
    }
    __syncthreads();

    float gM = NEGINF;
    #pragma unroll
    for (int w = 0; w < NWAVE; ++w) gM = fmaxf(gM, red_m[w * 16 + lh]);
    const float beta = __expf(m - gM);

    #pragma unroll
    for (int dt = 0; dt < 8; ++dt) {
        #pragma unroll
        for (int r = 0; r < 8; ++r) {
            const int d = dt * 16 + r + (upper ? 8 : 0);
            atomicAdd(&outbuf[d * 16 + lh], acc[dt][r] * beta);   // ds_add_f32
        }
    }
    __syncthreads();

    // ---- normalize + write out[b, 0, kv*G + h, d]
    for (int e = tid; e < GQ * DH; e += 256) {
        const int h = e / DH;      // 0..3
        const int d = e % DH;
        float gM2 = NEGINF;
        #pragma unroll
        for (int w = 0; w < NWAVE; ++w) gM2 = fmaxf(gM2, red_m[w * 16 + h]);
        float Ls = 0.0f;
        #pragma unroll
        for (int w = 0; w < NWAVE; ++w)
            Ls += red_l[w * 16 + h] * __expf(red_m[w * 16 + h] - gM2);
        out[((size_t)b * HQ + (size_t)kv * GQ + h) * DH + d] = outbuf[d * 16 + h] / Ls;
    }
}

extern "C" void kernel_launch(void* const* d_in, const int* in_sizes, int n_in,
                              void* d_out, int out_size, void* d_ws, size_t ws_size,
                              hipStream_t stream) {
    (void)in_sizes; (void)n_in; (void)d_ws; (void)ws_size; (void)out_size;
    const float* q         = (const float*)d_in[0];
    const float* k         = (const float*)d_in[1];
    const float* v         = (const float*)d_in[2];
    const float* k_cache   = (const float*)d_in[3];
    const float* v_cache   = (const float*)d_in[4];
    const int*   ctx_lens  = (const int*)d_in[5];
    const int*   btab      = (const int*)d_in[6];
    // d_in[7] = slot_mapping: not needed (fresh row substituted at ctx-1).
    float* out = (float*)d_out;

    dim3 grid(KVH, BQ);
    dim3 block(256);
    paged_attn_decode_kernel<<<grid, block, 0, stream>>>(
        q, k, v, k_cache, v_cache, ctx_lens, btab, out);
}